// Transition_28578712387757
// MI455X (gfx1250) — compile-verified
//
#include <hip/hip_runtime.h>
#include <math.h>

typedef float v2f __attribute__((ext_vector_type(2)));
typedef float v8f __attribute__((ext_vector_type(8)));

#define CB 2
#define CC 64
#define CN 16384
#define CM 2048
#define NSAMP 32
#define R2 0.01f
#define FPS_THREADS 1024
#define PPT (CN / FPS_THREADS)   // 16

// ---------------------------------------------------------------------------
// Kernel 1: conv1x1 as GEMM via V_WMMA_F32_16X16X4_F32.
// Each wave computes one 16(co) x 16(n) output tile, K-loop over C=64 in steps of 4.
// ---------------------------------------------------------------------------
__global__ __launch_bounds__(256) void conv_wmma_kernel(const float* __restrict__ x,
                                                        const float* __restrict__ W,
                                                        float* __restrict__ h) {
  const int ntiles = CN >> 4;                    // 1024 n-tiles per batch
  const int wavesPerBatch = ntiles * (CC >> 4);  // 4096
  int gw = blockIdx.x * 8 + (threadIdx.x >> 5);
  int lane = threadIdx.x & 31;
  int b = gw / wavesPerBatch;
  int r = gw % wavesPerBatch;
  int co = (r / ntiles) << 4;
  int n0 = (r % ntiles) << 4;
  int half = lane >> 4;
  int l = lane & 15;

  v8f c = {0.f, 0.f, 0.f, 0.f, 0.f, 0.f, 0.f, 0.f};
  const float* xb = x + (size_t)b * CC * CN;

  for (int kb = 0; kb < CC; kb += 4) {
    // A = W tile [16 x 4]: lane l holds row co+l; half selects K pair {0,1} / {2,3}
    const float* wrow = W + (size_t)(co + l) * CC + kb + 2 * half;
    v2f a;
    a.x = wrow[0];
    a.y = wrow[1];
    // B = x tile [4 x 16]: lane l holds column n0+l; half selects K pair
    const float* xcol = xb + (size_t)(kb + 2 * half) * CN + n0 + l;
    v2f bb;
    bb.x = xcol[0];
    bb.y = xcol[CN];
    c = __builtin_amdgcn_wmma_f32_16x16x4_f32(false, a, false, bb, (short)0, c,
                                              false, false);
  }

  float* hb = h + (size_t)b * CC * CN;
#pragma unroll
  for (int rr = 0; rr < 8; rr++)
    hb[(size_t)(co + rr + 8 * half) * CN + n0 + l] = c[rr];
}

// ---------------------------------------------------------------------------
// Kernel 2: per-channel batch stats -> fold into scale/bias.  One block/channel.
// ---------------------------------------------------------------------------
__global__ __launch_bounds__(256) void bn_stats_kernel(const float* __restrict__ h,
                                                       const float* __restrict__ gamma,
                                                       const float* __restrict__ beta,
                                                       float* __restrict__ scaleArr,
                                                       float* __restrict__ biasArr) {
  __shared__ float ss[256], sq[256];
  int cch = blockIdx.x;
  int tid = threadIdx.x;
  float s = 0.f, q = 0.f;
  for (int b = 0; b < CB; b++) {
    const float* hp = h + ((size_t)b * CC + cch) * CN;
    for (int i = tid; i < CN; i += 256) {
      float v = hp[i];
      s += v;
      q += v * v;
    }
  }
  ss[tid] = s;
  sq[tid] = q;
  __syncthreads();
  for (int o = 128; o > 0; o >>= 1) {
    if (tid < o) {
      ss[tid] += ss[tid + o];
      sq[tid] += sq[tid + o];
    }
    __syncthreads();
  }
  if (tid == 0) {
    const float inv = 1.0f / (float)(CB * CN);
    float mean = ss[0] * inv;
    float var = sq[0] * inv - mean * mean;
    float sc = gamma[cch] * rsqrtf(var + 1e-5f);
    scaleArr[cch] = sc;
    biasArr[cch] = beta[cch] - mean * sc;
  }
}

// ---------------------------------------------------------------------------
// Kernel 3: apply BN+ReLU and transpose [B,C,N] -> [B,N,C] via LDS 64x64 tile.
// ---------------------------------------------------------------------------
__global__ __launch_bounds__(256) void bn_apply_tr_kernel(const float* __restrict__ h,
                                                          const float* __restrict__ scaleArr,
                                                          const float* __restrict__ biasArr,
                                                          float* __restrict__ xt) {
  __shared__ float tile[64][65];
  int b = blockIdx.y;
  int n0 = blockIdx.x * 64;
  int tx = threadIdx.x & 63;
  int ty = threadIdx.x >> 6;  // 0..3
  const float* hb = h + (size_t)b * CC * CN;
#pragma unroll
  for (int i = 0; i < 16; i++) {
    int cch = ty * 16 + i;
    float v = hb[(size_t)cch * CN + n0 + tx];
    v = scaleArr[cch] * v + biasArr[cch];
    tile[cch][tx] = fmaxf(v, 0.0f);
  }
  __syncthreads();
  float* xb = xt + (size_t)b * CN * CC + (size_t)n0 * CC;
#pragma unroll
  for (int i = 0; i < 16; i++) {
    int n = ty * 16 + i;
    xb[(size_t)n * CC + tx] = tile[tx][n];
  }
}

// ---------------------------------------------------------------------------
// Kernel 4: furthest point sampling. One 1024-thread block per batch.
// Coords in registers for the hot distance loop; duplicated in 192KB dynamic
// LDS for winner-coordinate broadcast (CDNA5: 320KB LDS per WGP).
// Writes sampled coords directly into d_out region 0 ([B,3,M]).
// ---------------------------------------------------------------------------
__global__ __launch_bounds__(FPS_THREADS) void fps_kernel(const float* __restrict__ p,
                                                          float* __restrict__ out0) {
  extern __shared__ float smem[];
  float* spx = smem;
  float* spy = smem + CN;
  float* spz = smem + 2 * CN;
  __shared__ float wv[32];
  __shared__ int wi[32];
  __shared__ int sIdx;

  const int b = blockIdx.x;
  const int tid = threadIdx.x;
  const int lane = tid & 31;
  const int wid = tid >> 5;
  const float* pb = p + (size_t)b * 3 * CN;

  float px[PPT], py[PPT], pz[PPT], dist[PPT];
#pragma unroll
  for (int j = 0; j < PPT; j++) {
    int n = tid + j * FPS_THREADS;
    float xx = pb[n], yy = pb[CN + n], zz = pb[2 * CN + n];
    px[j] = xx; py[j] = yy; pz[j] = zz;
    dist[j] = 1e10f;
    spx[n] = xx; spy[n] = yy; spz[n] = zz;
  }
  if (tid == 0) {
    sIdx = 0;
    out0[(size_t)b * 3 * CM + 0 * CM + 0] = pb[0];
    out0[(size_t)b * 3 * CM + 1 * CM + 0] = pb[CN];
    out0[(size_t)b * 3 * CM + 2 * CM + 0] = pb[2 * CN];
  }
  __syncthreads();

  int win = 0;
  for (int m = 1; m < CM; m++) {
    float bx = spx[win], by = spy[win], bz = spz[win];
    float bv = -1.0f;
    int bi = 0;
#pragma unroll
    for (int j = 0; j < PPT; j++) {
      float dx = px[j] - bx, dy = py[j] - by, dz = pz[j] - bz;
      float d = dx * dx + dy * dy + dz * dz;
      float dm = fminf(dist[j], d);
      dist[j] = dm;
      if (dm > bv) { bv = dm; bi = j * FPS_THREADS + tid; }
    }
    // wave32 argmax reduce (lowest index wins ties)
#pragma unroll
    for (int off = 16; off > 0; off >>= 1) {
      float ov = __shfl_xor(bv, off, 32);
      int oi = __shfl_xor(bi, off, 32);
      if (ov > bv || (ov == bv && oi < bi)) { bv = ov; bi = oi; }
    }
    if (lane == 0) { wv[wid] = bv; wi[wid] = bi; }
    __syncthreads();
    if (wid == 0) {
      float v2 = wv[lane];
      int i2 = wi[lane];
#pragma unroll
      for (int off = 16; off > 0; off >>= 1) {
        float ov = __shfl_xor(v2, off, 32);
        int oi = __shfl_xor(i2, off, 32);
        if (ov > v2 || (ov == v2 && oi < i2)) { v2 = ov; i2 = oi; }
      }
      if (lane == 0) {
        sIdx = i2;
        out0[(size_t)b * 3 * CM + 0 * CM + m] = spx[i2];
        out0[(size_t)b * 3 * CM + 1 * CM + m] = spy[i2];
        out0[(size_t)b * 3 * CM + 2 * CM + m] = spz[i2];
      }
    }
    __syncthreads();
    win = sIdx;
  }
}

// ---------------------------------------------------------------------------
// Kernel 5: fused ball query + gather + channel max. One wave per (b, m).
// Lanes scan 32 points/step; ballot+prefix-popcount assigns ascending-index
// neighbor slots (reference ball-query semantics), pad with first hit.
// Then each lane maxes channels {lane, lane+32} over the 32 neighbors.
// ---------------------------------------------------------------------------
__global__ __launch_bounds__(256) void bqg_kernel(const float* __restrict__ p,
                                                  const float* __restrict__ q,
                                                  const float* __restrict__ xt,
                                                  float* __restrict__ out1) {
  __shared__ int nbr[8][NSAMP];
  int w = threadIdx.x >> 5;
  int lane = threadIdx.x & 31;
  int gw = blockIdx.x * 8 + w;
  int b = gw / CM;
  int m = gw % CM;

  const float* qb = q + (size_t)b * 3 * CM;
  float qx = qb[m], qy = qb[CM + m], qz = qb[2 * CM + m];

  const float* pb = p + (size_t)b * 3 * CN;
  int cnt = 0;
  for (int base = 0; base < CN && cnt < NSAMP; base += 32) {
    int n = base + lane;
    float dx = pb[n] - qx, dy = pb[CN + n] - qy, dz = pb[2 * CN + n] - qz;
    float d2 = dx * dx + dy * dy + dz * dz;
    bool in = d2 < R2;
    unsigned mask = (unsigned)__ballot(in);
    if (in) {
      int pos = cnt + __popc(mask & ((1u << lane) - 1));
      if (pos < NSAMP) nbr[w][pos] = n;
    }
    cnt += __popc(mask);
  }
  if (cnt > NSAMP) cnt = NSAMP;
  asm volatile("s_wait_dscnt 0" ::: "memory");
  int first = nbr[w][0];
  if (lane >= cnt) nbr[w][lane] = first;  // pad with first found index
  asm volatile("s_wait_dscnt 0" ::: "memory");

  const float* xb = xt + (size_t)b * CN * CC;
  float m0 = -1e30f, m1 = -1e30f;
#pragma unroll 4
  for (int j = 0; j < NSAMP; j++) {
    int nj = nbr[w][j];
    const float* row = xb + (size_t)nj * CC;
    m0 = fmaxf(m0, row[lane]);
    m1 = fmaxf(m1, row[lane + 32]);
  }
  float* ob = out1 + (size_t)b * CC * CM;
  ob[(size_t)lane * CM + m] = m0;
  ob[(size_t)(lane + 32) * CM + m] = m1;
}

// ---------------------------------------------------------------------------
extern "C" void kernel_launch(void* const* d_in, const int* in_sizes, int n_in,
                              void* d_out, int out_size, void* d_ws, size_t ws_size,
                              hipStream_t stream) {
  (void)in_sizes; (void)n_in; (void)out_size; (void)ws_size;
  const float* p     = (const float*)d_in[0];  // [B,3,N]
  const float* x     = (const float*)d_in[1];  // [B,C,N]
  const float* W     = (const float*)d_in[2];  // [C,C]
  const float* gamma = (const float*)d_in[3];  // [C]
  const float* beta  = (const float*)d_in[4];  // [C]

  float* ws = (float*)d_ws;
  float* h        = ws;                               // [B,C,N]
  float* xt       = h + (size_t)CB * CC * CN;         // [B,N,C]
  float* scaleArr = xt + (size_t)CB * CC * CN;        // [C]
  float* biasArr  = scaleArr + CC;                    // [C]

  float* out0 = (float*)d_out;                        // [B,3,M]
  float* out1 = out0 + (size_t)CB * 3 * CM;           // [B,C,M]

  // conv1x1 via WMMA
  conv_wmma_kernel<<<dim3((CB * (CC / 16) * (CN / 16)) / 8), 256, 0, stream>>>(x, W, h);
  // batch-norm stats (training mode)
  bn_stats_kernel<<<dim3(CC), 256, 0, stream>>>(h, gamma, beta, scaleArr, biasArr);
  // apply BN + ReLU, transpose to [B,N,C]
  bn_apply_tr_kernel<<<dim3(CN / 64, CB), 256, 0, stream>>>(h, scaleArr, biasArr, xt);
  // furthest point sampling (writes p_n into d_out region 0)
  fps_kernel<<<dim3(CB), FPS_THREADS, 3 * CN * sizeof(float), stream>>>(p, out0);
  // ball query + gather + max-pool (writes d_out region 1)
  bqg_kernel<<<dim3((CB * CM) / 8), 256, 0, stream>>>(p, out0, xt, out1);
}